// SimpleGCNLayer_13374528159917
// MI455X (gfx1250) — compile-verified
//
#include <hip/hip_runtime.h>

typedef __attribute__((ext_vector_type(2))) float v2f;
typedef __attribute__((ext_vector_type(8))) float v8f;

#define N_NODES   100000
#define DIM       128
#define N_EDGES   640000
#define LDS_STRIDE 132            // 132 % 64 == 4 -> conflict-free b64 B reads
#define N_TILES   (N_NODES / 16)  // 6250, exact

// ---------------------------------------------------------------------------
// Kernel 1: per-edge scatter. One wave32 per edge: each lane moves 4 contiguous
// channels (float4 load, 4x global_atomic_add_f32). Rows are 51.2MB total ->
// resident in 192MB L2, so atomics stay on-die.
// ---------------------------------------------------------------------------
__global__ __launch_bounds__(256) void gcn_scatter(
    const float* __restrict__ x,
    const int*   __restrict__ src,
    const int*   __restrict__ dst,
    float*       __restrict__ agg,
    float*       __restrict__ deg)
{
    int wave = (blockIdx.x * blockDim.x + threadIdx.x) >> 5;
    int lane = threadIdx.x & 31;
    if (wave >= N_EDGES) return;

    int s = src[wave];
    int d = dst[wave];

    const float4* xr = (const float4*)(x + (size_t)s * DIM);
    float*        ar = agg + (size_t)d * DIM;

    float4 v = xr[lane];                  // 32 lanes * 16B = full 512B row
    int c = lane << 2;
    atomicAdd(ar + c + 0, v.x);
    atomicAdd(ar + c + 1, v.y);
    atomicAdd(ar + c + 2, v.z);
    atomicAdd(ar + c + 3, v.w);
    if (lane == 0) atomicAdd(deg + d, 1.0f);
}

// ---------------------------------------------------------------------------
// Kernel 2: fused (agg/deg + x) @ W^T + b, ReLU, via V_WMMA_F32_16X16X4_F32.
// 8 waves/block, one 16-row node tile per wave, 8 N-tiles (full 128 cols).
// W cached in LDS with padded stride for bank-conflict-free B fragments.
// ---------------------------------------------------------------------------
__global__ __launch_bounds__(256) void gcn_gemm(
    const float* __restrict__ x,
    const float* __restrict__ agg,
    const float* __restrict__ deg,
    const float* __restrict__ W,      // [out=128][in=128] row-major
    const float* __restrict__ bias,   // [128]
    float*       __restrict__ out)    // [N][128]
{
    __shared__ float wlds[DIM * LDS_STRIDE];

    // Cooperative stage of W into LDS (float4 granularity, padded rows).
    for (int i = threadIdx.x; i < DIM * (DIM / 4); i += 256) {
        int o  = i >> 5;            // 32 float4 per row
        int k4 = (i & 31) << 2;
        float4 wv = *(const float4*)(W + o * DIM + k4);
        *(float4*)(&wlds[o * LDS_STRIDE + k4]) = wv;
    }
    __syncthreads();

    int wave = threadIdx.x >> 5;
    int lane = threadIdx.x & 31;
    int tile = blockIdx.x * 8 + wave;
    if (tile >= N_TILES) return;

    int m     = lane & 15;            // M row (A) / N col (B,C,D) within tile
    int khalf = (lane >> 4) << 1;     // lanes 0-15 -> K{0,1}; 16-31 -> K{2,3}
    int node  = tile * 16 + m;

    float dg   = deg[node];
    float invd = 1.0f / fmaxf(dg, 1.0f);

    v8f acc[8] = {};                  // 8 N-tiles x 8 VGPRs f32 accumulators

    const float* xrow = x   + (size_t)node * DIM;
    const float* arow = agg + (size_t)node * DIM;

    #pragma unroll
    for (int kk = 0; kk < DIM; kk += 4) {
        int k0 = kk + khalf;
        v2f xa = *(const v2f*)(xrow + k0);    // b64 global load
        v2f aa = *(const v2f*)(arow + k0);    // b64 global load
        v2f a  = aa * invd + xa;              // fused mean-agg + residual

        #pragma unroll
        for (int nt = 0; nt < 8; ++nt) {
            // B[k][n] = W[n][k]; lane holds col n = nt*16 + m, K-pair k0,k0+1
            v2f b = *(const v2f*)(&wlds[(nt * 16 + m) * LDS_STRIDE + k0]);
            acc[nt] = __builtin_amdgcn_wmma_f32_16x16x4_f32(
                false, a, false, b, (short)0, acc[nt], false, false);
        }
    }

    // Epilogue: C/D layout -> row = v + 8*(lane>=16), col = lane&15.
    int rbase = tile * 16 + ((lane >> 4) << 3);
    #pragma unroll
    for (int nt = 0; nt < 8; ++nt) {
        int col = nt * 16 + m;
        float bv = bias[col];
        #pragma unroll
        for (int v = 0; v < 8; ++v) {
            float r = fmaxf(acc[nt][v] + bv, 0.0f);
            out[(size_t)(rbase + v) * DIM + col] = r;
        }
    }
}

// ---------------------------------------------------------------------------
extern "C" void kernel_launch(void* const* d_in, const int* in_sizes, int n_in,
                              void* d_out, int out_size, void* d_ws, size_t ws_size,
                              hipStream_t stream) {
    const float* x    = (const float*)d_in[0];                 // [N,128]
    const int*   eidx = (const int*)  d_in[1];                 // [2,E]
    const float* W    = (const float*)d_in[2];                 // [128,128]
    const float* bias = (const float*)d_in[3];                 // [128]
    float*       out  = (float*)d_out;

    const int* src = eidx;                                     // row 0
    const int* dst = eidx + N_EDGES;                           // row 1

    float* agg = (float*)d_ws;                                 // N*128 f32
    float* deg = agg + (size_t)N_NODES * DIM;                  // N f32

    size_t zero_bytes = ((size_t)N_NODES * DIM + N_NODES) * sizeof(float);
    hipMemsetAsync(d_ws, 0, zero_bytes, stream);

    // One wave per edge: 8 edges per 256-thread block.
    int scatter_blocks = N_EDGES / 8;                          // 80000
    gcn_scatter<<<scatter_blocks, 256, 0, stream>>>(x, src, dst, agg, deg);

    // One 16-row tile per wave, 8 waves per block.
    int gemm_blocks = (N_TILES + 7) / 8;                       // 782
    gcn_gemm<<<gemm_blocks, 256, 0, stream>>>(x, agg, deg, W, bias, out);
}